// RegionLoss_3D_multi_three_61263413510186
// MI455X (gfx1250) — compile-verified
//
#include <hip/hip_runtime.h>
#include <math.h>

typedef __attribute__((ext_vector_type(2))) float v2f;
typedef __attribute__((ext_vector_type(8))) float v8f;

#define DIMS 96
#define RBLK 32          // region edge (96/3)
#define XS   33          // padded LDS row stride for X (32 cols)
#define GS   65          // padded LDS row stride for G (64 cols)

// ---------------------------------------------------------------------------
// Kernel 1: streaming block-sum. One workgroup per (tensor, b, c, region).
// Each region is 32^3 f32 = 128 KB; 256 threads x 32 float4 loads each.
// Bandwidth-bound: 453 MB total reads -> ~19.4 us floor at 23.3 TB/s.
// ---------------------------------------------------------------------------
__global__ void region_pool_kernel(const float* __restrict__ p0,
                                   const float* __restrict__ p1,
                                   float* __restrict__ means) {
    __shared__ float red[256];
    const int blk = blockIdx.x;            // 0..3455
    const int t   = blk / 1728;            // tensor select
    const int rem = blk % 1728;
    const int bc  = rem / 27;              // b*32 + c  (0..63)
    const int r   = rem % 27;              // region id (d,h,w order)
    const int bz  = r / 9;
    const int by  = (r % 9) / 3;
    const int bx  = r % 3;

    const float* __restrict__ src = (t == 0) ? p0 : p1;
    const size_t base = ((size_t)bc * DIMS + (size_t)(bz * RBLK)) * (DIMS * DIMS)
                      + (size_t)(by * RBLK) * DIMS + (size_t)(bx * RBLK);

    const int tid = threadIdx.x;
    float acc = 0.0f;
    // 8192 float4 per region; thread handles i = tid + 256*k (coalesced 128B rows)
    #pragma unroll 4
    for (int k = 0; k < 32; ++k) {
        const int i    = tid + (k << 8);
        const int z    = i >> 8;           // 0..31
        const int rest = i & 255;
        const int y    = rest >> 3;        // 0..31
        const int v    = rest & 7;         // float4 within row
        const size_t off = base + (size_t)z * (DIMS * DIMS) + (size_t)y * DIMS + (v << 2);
        const float4 d = *reinterpret_cast<const float4*>(src + off);
        acc += (d.x + d.y) + (d.z + d.w);
    }

    red[tid] = acc;
    __syncthreads();
    for (int s = 128; s > 0; s >>= 1) {
        if (tid < s) red[tid] += red[tid + s];
        __syncthreads();
    }
    if (tid == 0) means[blk] = red[0] * (1.0f / 32768.0f);
}

// ---------------------------------------------------------------------------
// Kernel 2: single workgroup finisher.
//   X[t] : 64x32 (rows 0..53 = region means b*27+r; rows 54,55 = pool1 means;
//          rows 56..63 = zero padding)
//   G[t] = X[t] X[t]^T via v_wmma_f32_16x16x4_f32 (32 waves, one 16x16 tile each)
//   diff1 from G rows/cols {54,55}; diff2=diff3 from rows/cols 0..53.
// ---------------------------------------------------------------------------
__global__ void cosine_mse_kernel(const float* __restrict__ means,
                                  float* __restrict__ out) {
    __shared__ float X[2][64 * XS];
    __shared__ float G[2][64 * GS];
    __shared__ float nrm[2][64];
    __shared__ float red[1024];

    const int tid = threadIdx.x;

    // zero X (padding rows matter)
    for (int i = tid; i < 2 * 64 * XS; i += 1024) (&X[0][0])[i] = 0.0f;
    __syncthreads();

    // rows 0..53: X[t][b*27+r][c] = means[t*1728 + (b*32+c)*27 + r]
    for (int i = tid; i < 2 * 54 * 32; i += 1024) {
        const int tt  = i / (54 * 32);
        const int rm  = i % (54 * 32);
        const int row = rm / 32;
        const int c   = rm % 32;
        const int b   = row / 27;
        const int r   = row % 27;
        X[tt][row * XS + c] = means[tt * 1728 + (b * 32 + c) * 27 + r];
    }
    __syncthreads();

    // rows 54,55: pool1 means = average of the 27 region means
    if (tid < 128) {
        const int tt = tid / 64;
        const int bc = tid % 64;
        const int b  = bc / 32;
        const int c  = bc % 32;
        float s = 0.0f;
        for (int r = 0; r < 27; ++r) s += X[tt][(b * 27 + r) * XS + c];
        X[tt][(54 + b) * XS + c] = s * (1.0f / 27.0f);
    }
    __syncthreads();

    // --- WMMA: G[t] = X[t] X[t]^T, 64x64, K=32 in 8 steps of K=4 ---
    {
        const int wave = tid >> 5;
        const int lane = tid & 31;
        const int t    = wave >> 4;          // tensor
        const int tile = wave & 15;          // 4x4 tile grid
        const int i0   = (tile >> 2) << 4;
        const int j0   = (tile & 3) << 4;
        const int m    = lane & 15;          // M (or N) within tile
        const int kk   = (lane >> 4) << 1;   // K sub-offset per lane half

        v8f c8 = {};
        #pragma unroll
        for (int k4 = 0; k4 < 32; k4 += 4) {
            v2f a, b;
            a.x = X[t][(i0 + m) * XS + k4 + kk];
            a.y = X[t][(i0 + m) * XS + k4 + kk + 1];
            b.x = X[t][(j0 + m) * XS + k4 + kk];
            b.y = X[t][(j0 + m) * XS + k4 + kk + 1];
            c8 = __builtin_amdgcn_wmma_f32_16x16x4_f32(
                    false, a, false, b, (short)0, c8, false, false);
        }
        // C/D layout: lanes 0-15 -> M=v, N=lane; lanes 16-31 -> M=v+8, N=lane-16
        const int rbase = i0 + ((lane >> 4) << 3);
        const int col   = j0 + (lane & 15);
        #pragma unroll
        for (int v = 0; v < 8; ++v) G[t][(rbase + v) * GS + col] = c8[v];
    }
    __syncthreads();

    // norms from Gram diagonal
    if (tid < 128) {
        const int tt = tid / 64;
        const int i  = tid % 64;
        nrm[tt][i] = sqrtf(G[tt][i * GS + i]);
    }
    __syncthreads();

    // diff2 (=diff3): 54x54 cosine MSE; diff1: 2x2 at rows/cols 54..55
    float acc2 = 0.0f;
    for (int i = tid; i < 54 * 54; i += 1024) {
        const int ri = i / 54, rj = i % 54;
        const float s1 = G[0][ri * GS + rj] / fmaxf(nrm[0][ri] * nrm[0][rj], 1e-8f);
        const float s2 = G[1][ri * GS + rj] / fmaxf(nrm[1][ri] * nrm[1][rj], 1e-8f);
        const float d  = s1 - s2;
        acc2 += d * d;
    }
    float acc1 = 0.0f;
    if (tid < 4) {
        const int ri = 54 + (tid >> 1), rj = 54 + (tid & 1);
        const float s1 = G[0][ri * GS + rj] / fmaxf(nrm[0][ri] * nrm[0][rj], 1e-8f);
        const float s2 = G[1][ri * GS + rj] / fmaxf(nrm[1][ri] * nrm[1][rj], 1e-8f);
        const float d  = s1 - s2;
        acc1 = d * d;
    }
    // per-thread partial of (mse1 + 2*mse2)
    red[tid] = acc1 * (1.0f / 4.0f) + acc2 * (2.0f / (54.0f * 54.0f));
    __syncthreads();
    for (int s = 512; s > 0; s >>= 1) {
        if (tid < s) red[tid] += red[tid + s];
        __syncthreads();
    }
    if (tid == 0) out[0] = red[0] * (1.0f / 3.0f);
}

extern "C" void kernel_launch(void* const* d_in, const int* in_sizes, int n_in,
                              void* d_out, int out_size, void* d_ws, size_t ws_size,
                              hipStream_t stream) {
    (void)in_sizes; (void)n_in; (void)out_size; (void)ws_size;
    const float* p1 = (const float*)d_in[0];
    const float* p2 = (const float*)d_in[1];
    float* means = (float*)d_ws;           // 2 * 2 * 32 * 27 floats = 13.5 KB

    region_pool_kernel<<<dim3(3456), dim3(256), 0, stream>>>(p1, p2, means);
    cosine_mse_kernel<<<dim3(1), dim3(1024), 0, stream>>>(means, (float*)d_out);
}